// WindowAttention_19301583028760
// MI455X (gfx1250) — compile-verified
//
#include <hip/hip_runtime.h>
#include <hip/hip_bf16.h>

typedef float v2f __attribute__((ext_vector_type(2)));
typedef float v8f __attribute__((ext_vector_type(8)));

#define SCALE_Q 0.17677669529663687f   // 1/sqrt(32)

// ---------------------------------------------------------------------------
// Kernel 1: relative-position bias MLP (depends only on window geometry).
// bias_g layout: [head][n][m], 3*64*64 floats in d_ws.
// ---------------------------------------------------------------------------
__global__ __launch_bounds__(256)
void bias_mlp_kernel(const float* __restrict__ w1, const float* __restrict__ b1,
                     const float* __restrict__ w2, const float* __restrict__ b2,
                     float* __restrict__ bias_g) {
  int t = blockIdx.x * 256 + threadIdx.x;      // 0..4095 = (n,m) pair
  int n = t >> 6, m = t & 63;
  int ny = n >> 3, nx = n & 7, my = m >> 3, mx = m & 7;
  float d0 = (float)(ny - my), d1 = (float)(nx - mx);
  float r0 = copysignf(log1pf(fabsf(d0)), d0);
  float r1 = copysignf(log1pf(fabsf(d1)), d1);
  float o0 = b2[0], o1 = b2[1], o2 = b2[2];
  #pragma unroll 8
  for (int u = 0; u < 256; ++u) {
    float hsum = fmaf(r0, w1[u], fmaf(r1, w1[256 + u], b1[u]));
    hsum = fmaxf(hsum, 0.0f);
    o0 = fmaf(hsum, w2[u * 3 + 0], o0);
    o1 = fmaf(hsum, w2[u * 3 + 1], o1);
    o2 = fmaf(hsum, w2[u * 3 + 2], o2);
  }
  int idx = n * 64 + m;
  bias_g[idx]        = o0;
  bias_g[4096 + idx] = o1;
  bias_g[8192 + idx] = o2;
}

// ---------------------------------------------------------------------------
// Kernel 2: one workgroup (256 threads = 8 wave32) per (batch, head).
//   load q,k,v -> LDS; circular 2D conv (VALU); S = conv(q,k) @ v^T + bias
//   via v_wmma_f32_16x16x4_f32; softmax; X = P @ v via WMMA; store.
// ---------------------------------------------------------------------------
__global__ __launch_bounds__(256)
void window_attn_kernel(const float* __restrict__ qkv,
                        const float* __restrict__ bias_g,
                        float* __restrict__ out_g) {
  __shared__ float qs[64][34];
  __shared__ float ks[64][34];
  __shared__ float vs[64][34];
  __shared__ float os[64][34];
  __shared__ float Ss[64][66];
  __shared__ float rinv[64];

  const int bh   = blockIdx.x;
  const int b    = bh / 3;
  const int head = bh % 3;
  const int tid  = threadIdx.x;
  const int w    = tid >> 5;          // wave id 0..7
  const int L    = tid & 31;          // lane
  const int row  = L & 15;
  const int hi   = L >> 4;            // half-wave select

  // ---- load q (pre-scaled), k, v : each [64 tok][32 ch] ----
  {
    const float* base = qkv + (size_t)b * 64 * 288 + head * 32;
    #pragma unroll
    for (int i = 0; i < 8; ++i) {
      int flat = tid + i * 256;             // 0..2047
      int n = flat >> 5, ch = flat & 31;
      const float* p = base + n * 288 + ch;
      qs[n][ch] = p[0] * SCALE_Q;
      ks[n][ch] = p[96];
      vs[n][ch] = p[192];
    }
  }
  __syncthreads();

  // ---- 2D circular convolution on 8x8 grid, per channel ----
  // wave w owns output rows n = 8w..8w+7 (ny == w); lane = channel.
  {
    const int ch = L;
    float acc[8] = {0,0,0,0,0,0,0,0};
    #pragma unroll
    for (int my = 0; my < 8; ++my) {
      int krow = ((w - my) & 7) * 8;
      float kreg[8];
      #pragma unroll
      for (int x = 0; x < 8; ++x) kreg[x] = ks[krow + x][ch];
      #pragma unroll
      for (int mx = 0; mx < 8; ++mx) {
        float qv = qs[my * 8 + mx][ch];
        #pragma unroll
        for (int j = 0; j < 8; ++j)
          acc[j] = fmaf(qv, kreg[(j - mx) & 7], acc[j]);
      }
    }
    #pragma unroll
    for (int j = 0; j < 8; ++j) os[w * 8 + j][ch] = acc[j];
  }
  __syncthreads();

  // ---- S[64,64] = conv_out[64,32] @ v^T[32,64] + bias  (WMMA f32) ----
  {
    const float* bias_h = bias_g + head * 4096;
    const int tn = w >> 1;
    const int n0 = tn * 16;
    #pragma unroll
    for (int tt = 0; tt < 2; ++tt) {
      const int m0 = ((w & 1) * 2 + tt) * 16;
      v8f c;
      #pragma unroll
      for (int r = 0; r < 8; ++r)
        c[r] = bias_h[(n0 + r + 8 * hi) * 64 + (m0 + row)];
      #pragma unroll
      for (int kq = 0; kq < 8; ++kq) {
        const int col = kq * 4 + 2 * hi;
        v2f a, bb;
        a.x  = os[n0 + row][col];  a.y  = os[n0 + row][col + 1];
        bb.x = vs[m0 + row][col];  bb.y = vs[m0 + row][col + 1];
        c = __builtin_amdgcn_wmma_f32_16x16x4_f32(
            false, a, false, bb, (short)0, c, false, false);
      }
      #pragma unroll
      for (int r = 0; r < 8; ++r)
        Ss[n0 + r + 8 * hi][m0 + row] = c[r];
    }
  }
  __syncthreads();

  // ---- row softmax (unnormalized exp in LDS, 1/sum deferred) ----
  if (tid < 64) {
    float mx = -3.4e38f;
    #pragma unroll 8
    for (int j = 0; j < 64; ++j) mx = fmaxf(mx, Ss[tid][j]);
    float sum = 0.0f;
    #pragma unroll 8
    for (int j = 0; j < 64; ++j) {
      float e = __expf(Ss[tid][j] - mx);
      Ss[tid][j] = e;
      sum += e;
    }
    rinv[tid] = 1.0f / sum;
  }
  __syncthreads();

  // ---- X[64,32] = P[64,64] @ v[64,32]  (WMMA f32), scale by 1/rowsum ----
  {
    const int n0 = (w >> 1) * 16;
    const int m0 = (w & 1) * 16;
    v8f c = {0.f, 0.f, 0.f, 0.f, 0.f, 0.f, 0.f, 0.f};
    #pragma unroll
    for (int kq = 0; kq < 16; ++kq) {
      const int col = kq * 4 + 2 * hi;
      v2f a, bb;
      a.x  = Ss[n0 + row][col];     a.y  = Ss[n0 + row][col + 1];
      bb.x = vs[col][m0 + row];     bb.y = vs[col + 1][m0 + row];
      c = __builtin_amdgcn_wmma_f32_16x16x4_f32(
          false, a, false, bb, (short)0, c, false, false);
    }
    #pragma unroll
    for (int r = 0; r < 8; ++r) {
      const int n = n0 + r + 8 * hi;
      out_g[((size_t)b * 64 + n) * 96 + head * 32 + m0 + row] = c[r] * rinv[n];
    }
  }
}

// ---------------------------------------------------------------------------
extern "C" void kernel_launch(void* const* d_in, const int* in_sizes, int n_in,
                              void* d_out, int out_size, void* d_ws, size_t ws_size,
                              hipStream_t stream) {
  const float* qkv = (const float*)d_in[0];
  const float* w1  = (const float*)d_in[1];
  const float* b1  = (const float*)d_in[2];
  const float* w2  = (const float*)d_in[3];
  const float* b2  = (const float*)d_in[4];
  float* out    = (float*)d_out;
  float* bias_g = (float*)d_ws;           // 3*64*64 floats

  bias_mlp_kernel<<<16, 256, 0, stream>>>(w1, b1, w2, b2, bias_g);
  window_attn_kernel<<<2048 * 3, 256, 0, stream>>>(qkv, bias_g, out);
}